// SimpleVisionTransformer_62234076119752
// MI455X (gfx1250) — compile-verified
//
#include <hip/hip_runtime.h>

#define IMG   144
#define NPS   36      // patches per side
#define NTOK  1296    // tokens per batch
#define EDIM  16
#define NBLK  81      // NTOK / 16
#define BATCH 4

typedef float v2f __attribute__((ext_vector_type(2)));
typedef float v8f __attribute__((ext_vector_type(8)));

// ---------------- workspace init: zero the max slot ----------------
__global__ void k_init(unsigned* m) { *m = 0u; }

// ---------------- global max over all 82944 elements ----------------
// launched with exactly BATCH*IMG*IMG threads; values are >= 0 so the
// uint bit pattern is order-preserving.
__global__ void k_max(const float* __restrict__ x, unsigned* __restrict__ m) {
  int tid = blockIdx.x * blockDim.x + threadIdx.x;
  float v = x[tid];
#pragma unroll
  for (int s = 16; s > 0; s >>= 1)
    v = fmaxf(v, __shfl_xor(v, s, 32));
  if ((threadIdx.x & 31) == 0)
    atomicMax(m, __float_as_uint(v));
}

// ---------------- per-token sum of squares (raw, un-normalized) ----------------
__global__ void k_sumsq(const float* __restrict__ x, float* __restrict__ sq) {
  int t = blockIdx.x * blockDim.x + threadIdx.x;
  if (t >= BATCH * NTOK) return;
  const float4* p = (const float4*)(x + (size_t)t * EDIM);
  float s = 0.f;
#pragma unroll
  for (int i = 0; i < 4; ++i) {
    float4 q = p[i];
    s += q.x * q.x + q.y * q.y + q.z * q.z + q.w * q.w;
  }
  sq[t] = s;
}

// ---------------- fused attention: one wave per 16-row i-block ----------------
// logits L[i,j] = (2*G[i,j] - |x_j|^2) / (16*x_max^2)   (row-const term dropped,
// softmax shift-invariant; L is bounded in [-1,2] so no running max needed)
// out rows = (sum_j exp(L) * X_j) / (sum_j exp(L))      (x_max cancels)
// exp folded to base-2: exp(L) = exp2(S*c1 - scol*c2e), 3 VALU/element.
__global__ void __launch_bounds__(32)
vit_main(const float* __restrict__ x, float* __restrict__ out,
         const unsigned* __restrict__ wmax, const float* __restrict__ sq)
{
  __shared__ float plds[256];            // P^T tile: plds[col*16 + row]
  const int lane = threadIdx.x;          // 0..31
  const int lo   = lane & 15;
  const int hi   = lane >> 4;            // 0 | 1
  const int iblk = blockIdx.x;           // 0..80
  const int b    = blockIdx.y;           // 0..3

  const float xm  = __uint_as_float(*wmax);
  const float inv = 1.0f / (16.0f * xm * xm);
  const float c2e = inv * 1.4426950408889634f;   // inv * log2(e)
  const float c1  = 2.0f * c2e;

  const float* xb  = x  + (size_t)b * NTOK * EDIM;
  const float* sqb = sq + b * NTOK;

  // A-operand chunks of X_i (16x16 tile, K split 4x4).
  // 32-bit A 16x4 layout: lanes 0-15 hold K=4c,4c+1 ; lanes 16-31 hold K=4c+2,4c+3.
  const float* atile = xb + iblk * 256;
  v2f A[4];
#pragma unroll
  for (int c = 0; c < 4; ++c)
    A[c] = *(const v2f*)(atile + lo * 16 + 4 * c + 2 * hi);

  v8f   O = {0.f, 0.f, 0.f, 0.f, 0.f, 0.f, 0.f, 0.f};   // numerator accumulator
  float lsum[8] = {0.f, 0.f, 0.f, 0.f, 0.f, 0.f, 0.f, 0.f}; // per-lane partial denoms

  for (int jb = 0; jb < NBLK; ++jb) {
    const float* btile = xb + jb * 256;

    // ---- GEMM 1: S = X_i * X_j^T  (B = X_j^T has identical lane layout to A) ----
    v8f S = {0.f, 0.f, 0.f, 0.f, 0.f, 0.f, 0.f, 0.f};
#pragma unroll
    for (int c = 0; c < 4; ++c) {
      v2f Bv = *(const v2f*)(btile + lo * 16 + 4 * c + 2 * hi);
      S = __builtin_amdgcn_wmma_f32_16x16x4_f32(false, A[c], false, Bv,
                                                (short)0, S, false, false);
    }

    // ---- logits -> exp2; C-layout row = r + 8*hi, col = lo ----
    const float off = sqb[jb * 16 + lo] * c2e;
    float p8[8];
#pragma unroll
    for (int r = 0; r < 8; ++r) {
      float e = __builtin_amdgcn_exp2f(__builtin_fmaf(S[r], c1, -off));
      p8[r] = e;
      lsum[r] += e;                                  // same row id as O[r]
    }

    // ---- transpose P through LDS into A-operand layout ----
    __syncthreads();                                 // WAR vs previous loads
    {
      float4* dst = (float4*)&plds[lo * 16 + 8 * hi];
      dst[0] = make_float4(p8[0], p8[1], p8[2], p8[3]);
      dst[1] = make_float4(p8[4], p8[5], p8[6], p8[7]);
    }
    __syncthreads();

    // ---- GEMM 2: O += P * X_j ----
#pragma unroll
    for (int c = 0; c < 4; ++c) {
      const int k = 4 * c + 2 * hi;
      v2f Pa = { plds[k * 16 + lo], plds[(k + 1) * 16 + lo] };
      // B = X_j (token-major): comp0 = X[4c+2hi][lo], comp1 = X[4c+1+2hi][lo]
      v2f Bx = { btile[(4 * c + 2 * hi) * 16 + lo],
                 btile[(4 * c + 1 + 2 * hi) * 16 + lo] };
      O = __builtin_amdgcn_wmma_f32_16x16x4_f32(false, Pa, false, Bx,
                                                (short)0, O, false, false);
    }
  }

  // ---- finalize denominators: reduce across the 16-lane half ----
#pragma unroll
  for (int r = 0; r < 8; ++r) {
    float v = lsum[r];
    v += __shfl_xor(v, 1, 16);
    v += __shfl_xor(v, 2, 16);
    v += __shfl_xor(v, 4, 16);
    v += __shfl_xor(v, 8, 16);
    lsum[r] = v;
  }

  // ---- normalize + fold permutation: img[by*4+ky][bx*4+kx] = out[n][d] ----
  float* ob = out + (size_t)b * IMG * IMG;
  const int ky = lo >> 2, kx = lo & 3;
#pragma unroll
  for (int r = 0; r < 8; ++r) {
    int n  = iblk * 16 + r + 8 * hi;
    int by = n / NPS, bx = n % NPS;
    ob[(by * 4 + ky) * IMG + bx * 4 + kx] = O[r] / lsum[r];
  }
}

extern "C" void kernel_launch(void* const* d_in, const int* in_sizes, int n_in,
                              void* d_out, int out_size, void* d_ws, size_t ws_size,
                              hipStream_t stream) {
  (void)in_sizes; (void)n_in; (void)out_size; (void)ws_size;
  const float* x = (const float*)d_in[0];
  float* out = (float*)d_out;
  unsigned* wmax = (unsigned*)d_ws;
  float* sq = (float*)d_ws + 16;          // 64-byte offset; needs 64 + 5184*4 bytes

  k_init<<<1, 1, 0, stream>>>(wmax);
  k_max<<<BATCH * IMG * IMG / 256, 256, 0, stream>>>(x, wmax);
  k_sumsq<<<(BATCH * NTOK + 127) / 128, 128, 0, stream>>>(x, sq);

  dim3 grid(NBLK, BATCH);                  // 81 x 4 single-wave workgroups
  vit_main<<<grid, 32, 0, stream>>>(x, out, wmax, sq);
}